// LinkPredictionGNN_43087111914333
// MI455X (gfx1250) — compile-verified
//
#include <hip/hip_runtime.h>
#include <hip/hip_bf16.h>

typedef __attribute__((ext_vector_type(2))) float v2f;
typedef __attribute__((ext_vector_type(8))) float v8f;

#define NFEAT 128

// ---------------------------------------------------------------------------
// Degree accumulation: deg[dst[e]] += 1
// ---------------------------------------------------------------------------
__global__ void __launch_bounds__(256) degree_kernel(const int* __restrict__ dstI,
                                                     float* __restrict__ deg, int E) {
  int e = blockIdx.x * blockDim.x + threadIdx.x;
  if (e < E) atomicAdd(&deg[dstI[e]], 1.0f);
}

// deg -> rsqrt(deg + 1)   (+1 for the implicit self loop)
__global__ void __launch_bounds__(256) dinv_kernel(float* __restrict__ deg, int N) {
  int i = blockIdx.x * blockDim.x + threadIdx.x;
  if (i < N) {
    float d = deg[i] + 1.0f;
    deg[i] = rsqrtf(d);
  }
}

// ---------------------------------------------------------------------------
// Pack a [K x 128] row-major weight matrix into K-pair-major float2 layout:
//   Wp[j*128 + col] = { W[2j][col], W[(2j+1)][col] }
// so a WMMA B-fragment is ONE aligned 64-bit load per lane.
// ---------------------------------------------------------------------------
__global__ void __launch_bounds__(256) pack_w_kernel(const float* __restrict__ W,
                                                     v2f* __restrict__ Wp, int Khalf) {
  int idx = blockIdx.x * blockDim.x + threadIdx.x;
  if (idx >= Khalf * NFEAT) return;
  int j = idx >> 7;
  int col = idx & (NFEAT - 1);
  v2f t;
  t.x = W[(size_t)(2 * j) * NFEAT + col];
  t.y = W[(size_t)(2 * j + 1) * NFEAT + col];
  Wp[idx] = t;
}

// ---------------------------------------------------------------------------
// C[M x 128] = A[M x 128] @ W[128 x 128] using V_WMMA_F32_16X16X4_F32.
// One wave per 16-row block, 8 waves per workgroup share W staged in LDS
// in K-pair-major float2 form (exactly 64 KB) -> single ds_load_b64 per frag.
// ---------------------------------------------------------------------------
__global__ void __launch_bounds__(256) gemm_node_kernel(const float* __restrict__ A,
                                                        const float* __restrict__ W,
                                                        float* __restrict__ C, int M) {
  __shared__ v2f Wp[(NFEAT / 2) * NFEAT];  // 64 * 128 * 8B = 64 KB
  for (int idx = threadIdx.x; idx < (NFEAT / 2) * NFEAT; idx += blockDim.x) {
    int j = idx >> 7;
    int col = idx & (NFEAT - 1);
    v2f t;
    t.x = W[(size_t)(2 * j) * NFEAT + col];
    t.y = W[(size_t)(2 * j + 1) * NFEAT + col];
    Wp[idx] = t;
  }
  __syncthreads();

  const int wave = threadIdx.x >> 5;   // 0..7, uniform per wave
  const int lane = threadIdx.x & 31;
  const int l16 = lane & 15;
  const int hi = lane >> 4;            // 0: K pair {k,k+1}, 1: {k+2,k+3}
  const int row0 = (blockIdx.x * 8 + wave) * 16;
  if (row0 >= M) return;               // wave-uniform exit: EXEC stays all-ones

  const v8f zero = {0.f, 0.f, 0.f, 0.f, 0.f, 0.f, 0.f, 0.f};
  v8f acc[8];
#pragma unroll
  for (int i = 0; i < 8; ++i) acc[i] = zero;

  int arow_idx = row0 + l16;
  if (arow_idx >= M) arow_idx = M - 1;           // clamp (writes guarded)
  const float* arow = A + (size_t)arow_idx * NFEAT;

  for (int k = 0; k < NFEAT; k += 4) {
    v2f a = *(const v2f*)(arow + k + hi * 2);                 // global b64
    const v2f* wrow = &Wp[(size_t)((k >> 1) + hi) * NFEAT + l16];
#pragma unroll
    for (int nt = 0; nt < 8; ++nt) {
      v2f b = wrow[nt * 16];                                  // ds_load_b64
      acc[nt] = __builtin_amdgcn_wmma_f32_16x16x4_f32(false, a, false, b, (short)0,
                                                      acc[nt], false, false);
    }
  }

  // C/D layout: VGPR i -> row (hi*8 + i), col = nt*16 + l16
#pragma unroll
  for (int i = 0; i < 8; ++i) {
    int r = row0 + hi * 8 + i;
    if (r < M) {
      float* crow = C + (size_t)r * NFEAT + l16;
#pragma unroll
      for (int nt = 0; nt < 8; ++nt) crow[nt * 16] = acc[nt][i];
    }
  }
}

// ---------------------------------------------------------------------------
// Edge scatter: agg[dst] += Hraw[src] * dinv[src]*dinv[dst]
// One wave per edge, one float4 per lane (32*16B = 512B row).
// ---------------------------------------------------------------------------
__global__ void __launch_bounds__(256) scatter_kernel(const float* __restrict__ H,
                                                      const int* __restrict__ srcI,
                                                      const int* __restrict__ dstI,
                                                      const float* __restrict__ dinv,
                                                      float* __restrict__ agg, int E) {
  long long gid = (long long)blockIdx.x * blockDim.x + threadIdx.x;
  int edge = (int)(gid >> 5);
  int lane = (int)(gid & 31);
  if (edge >= E) return;
  int s = srcI[edge];
  int d = dstI[edge];
  float norm = dinv[s] * dinv[d];
  float4 v = ((const float4*)(H + (size_t)s * NFEAT))[lane];
  float* out = agg + (size_t)d * NFEAT + lane * 4;
  atomicAdd(out + 0, v.x * norm);
  atomicAdd(out + 1, v.y * norm);
  atomicAdd(out + 2, v.z * norm);
  atomicAdd(out + 3, v.w * norm);
}

// ---------------------------------------------------------------------------
// Finalize: h = relu(agg + Hraw * dinv^2 + bias)   (self-loop fused, in place)
// ---------------------------------------------------------------------------
__global__ void __launch_bounds__(256) finalize_kernel(float* __restrict__ agg,
                                                       const float* __restrict__ Hraw,
                                                       const float* __restrict__ dinv,
                                                       const float* __restrict__ bias,
                                                       int N) {
  int gid = blockIdx.x * blockDim.x + threadIdx.x;
  if (gid >= N * NFEAT) return;
  int node = gid >> 7;
  int f = gid & (NFEAT - 1);
  float di = dinv[node];
  float v = agg[gid] + Hraw[gid] * (di * di) + bias[f];
  agg[gid] = fmaxf(v, 0.0f);
}

// ---------------------------------------------------------------------------
// Pair MLP: out[p] = relu([h[p0],h[p1]] @ Wh1 + bh1) @ Wh2 + bh2
// One wave per 16-pair block, K = 256 in WMMA steps of 4. Wh1 comes pre-packed
// (K-pair-major float2, 128 KB, L2-resident) -> one global b64 per fragment.
// Fused ReLU + 128->1 dot with shfl_xor reduction; no [P,128] intermediate.
// ---------------------------------------------------------------------------
__global__ void __launch_bounds__(256) pair_mlp_kernel(const float* __restrict__ H,
                                                       const int* __restrict__ pairs,
                                                       const v2f* __restrict__ Wh1p,
                                                       const float* __restrict__ bh1,
                                                       const float* __restrict__ Wh2,
                                                       const float* __restrict__ bh2,
                                                       float* __restrict__ out, int P) {
  const int wave = threadIdx.x >> 5;
  const int lane = threadIdx.x & 31;
  const int l16 = lane & 15;
  const int hi = lane >> 4;
  const int p0 = (blockIdx.x * 8 + wave) * 16;
  if (p0 >= P) return;                 // wave-uniform

  int pidx = p0 + l16;
  if (pidx >= P) pidx = P - 1;         // clamp; stores guarded
  const int pa = pairs[2 * pidx + 0];
  const int pb = pairs[2 * pidx + 1];
  const float* rowA = H + (size_t)pa * NFEAT;
  const float* rowB = H + (size_t)pb * NFEAT;

  const v8f zero = {0.f, 0.f, 0.f, 0.f, 0.f, 0.f, 0.f, 0.f};
  v8f acc[8];
#pragma unroll
  for (int i = 0; i < 8; ++i) acc[i] = zero;

  for (int k = 0; k < 2 * NFEAT; k += 4) {
    const float* r = (k < NFEAT) ? (rowA + k) : (rowB + (k - NFEAT));
    v2f a = *(const v2f*)(r + hi * 2);                          // global b64
    const v2f* wrow = &Wh1p[(size_t)((k >> 1) + hi) * NFEAT + l16];
#pragma unroll
    for (int nt = 0; nt < 8; ++nt) {
      v2f b = wrow[nt * 16];                                    // global b64 (L2)
      acc[nt] = __builtin_amdgcn_wmma_f32_16x16x4_f32(false, a, false, b, (short)0,
                                                      acc[nt], false, false);
    }
  }

  const float bias2 = bh2[0];
#pragma unroll
  for (int i = 0; i < 8; ++i) {
    float partial = 0.0f;
#pragma unroll
    for (int nt = 0; nt < 8; ++nt) {
      int col = nt * 16 + l16;
      float z = acc[nt][i] + bh1[col];
      z = fmaxf(z, 0.0f);
      partial += z * Wh2[col];
    }
    // reduce across the 16 lanes of this half (pair = p0 + hi*8 + i)
    partial += __shfl_xor(partial, 1, 32);
    partial += __shfl_xor(partial, 2, 32);
    partial += __shfl_xor(partial, 4, 32);
    partial += __shfl_xor(partial, 8, 32);
    int pr = p0 + hi * 8 + i;
    if (l16 == 0 && pr < P) out[pr] = partial + bias2;
  }
}

// ---------------------------------------------------------------------------
extern "C" void kernel_launch(void* const* d_in, const int* in_sizes, int n_in,
                              void* d_out, int out_size, void* d_ws, size_t ws_size,
                              hipStream_t stream) {
  const float* x = (const float*)d_in[0];
  const int* ei = (const int*)d_in[1];
  const int* pairs = (const int*)d_in[2];
  const float* W1 = (const float*)d_in[3];
  const float* b1 = (const float*)d_in[4];
  const float* W2 = (const float*)d_in[5];
  const float* b2 = (const float*)d_in[6];
  const float* Wh1 = (const float*)d_in[7];
  const float* bh1 = (const float*)d_in[8];
  const float* Wh2 = (const float*)d_in[9];
  const float* bh2 = (const float*)d_in[10];

  const int N = in_sizes[0] / NFEAT;
  const int E = in_sizes[1] / 2;
  const int P = in_sizes[2] / 2;
  const int* srcI = ei;        // edge_index row 0
  const int* dstI = ei + E;    // edge_index row 1

  // Workspace: dinv[N] | bufA[N*128] | bufB[N*128] | Wh1p[128*128 float2]
  char* ws = (char*)d_ws;
  size_t off = 0;
  auto alloc = [&](size_t bytes) -> void* {
    void* p = ws + off;
    off += (bytes + 255) & ~(size_t)255;
    return p;
  };
  float* dinv = (float*)alloc((size_t)N * sizeof(float));
  float* bufA = (float*)alloc((size_t)N * NFEAT * sizeof(float));
  float* bufB = (float*)alloc((size_t)N * NFEAT * sizeof(float));
  v2f* Wh1p = (v2f*)alloc((size_t)NFEAT * NFEAT * sizeof(v2f));  // K=256 -> 128 pairs

  // Degrees -> dinv ; pack Wh1 (independent, overlaps fine on one stream)
  hipMemsetAsync(dinv, 0, (size_t)N * sizeof(float), stream);
  degree_kernel<<<(E + 255) / 256, 256, 0, stream>>>(dstI, dinv, E);
  dinv_kernel<<<(N + 255) / 256, 256, 0, stream>>>(dinv, N);
  pack_w_kernel<<<(NFEAT * NFEAT + 255) / 256, 256, 0, stream>>>(Wh1, Wh1p, NFEAT);

  const int rowBlk = (N + 15) / 16;
  const int gemmGrid = (rowBlk + 7) / 8;
  const int scatGrid = (int)(((long long)E * 32 + 255) / 256);
  const int finGrid = (N * NFEAT + 255) / 256;

  // ---- Layer 1: bufA = x@W1 ; bufB = relu(scatter + selfloop + b1) ----
  gemm_node_kernel<<<gemmGrid, 256, 0, stream>>>(x, W1, bufA, N);
  hipMemsetAsync(bufB, 0, (size_t)N * NFEAT * sizeof(float), stream);
  scatter_kernel<<<scatGrid, 256, 0, stream>>>(bufA, srcI, dstI, dinv, bufB, E);
  finalize_kernel<<<finGrid, 256, 0, stream>>>(bufB, bufA, dinv, b1, N);

  // ---- Layer 2: bufA = h1@W2 ; bufB = relu(scatter + selfloop + b2) ----
  gemm_node_kernel<<<gemmGrid, 256, 0, stream>>>(bufB, W2, bufA, N);
  hipMemsetAsync(bufB, 0, (size_t)N * NFEAT * sizeof(float), stream);
  scatter_kernel<<<scatGrid, 256, 0, stream>>>(bufA, srcI, dstI, dinv, bufB, E);
  finalize_kernel<<<finGrid, 256, 0, stream>>>(bufB, bufA, dinv, b2, N);

  // ---- Pair MLP (fused) ----
  const int pBlk = (P + 15) / 16;
  pair_mlp_kernel<<<(pBlk + 7) / 8, 256, 0, stream>>>(bufB, pairs, Wh1p, bh1, Wh2, bh2,
                                                      (float*)d_out, P);
}